// RelationAwareAttention_11656541241625
// MI455X (gfx1250) — compile-verified
//
#include <hip/hip_runtime.h>

typedef float v2f __attribute__((ext_vector_type(2)));
typedef float v8f __attribute__((ext_vector_type(8)));

#define NH 8
#define LSEQ 512
#define DHEAD 64
#define TI 16          // query rows per workgroup
#define QS 72          // padded LDS stride for Q tile (floats, 16B-aligned rows)
#define SS 516         // padded LDS stride for score rows (floats)

__launch_bounds__(256)
__global__ void ra_attn_kernel(const float* __restrict__ q,
                               const float* __restrict__ kk_,
                               const float* __restrict__ v,
                               const float* __restrict__ rk,
                               const float* __restrict__ rv,
                               const int*   __restrict__ mask,
                               float* __restrict__ out_attn,
                               float* __restrict__ out_p)
{
    __shared__ float Qs[TI * QS];
    __shared__ float Sc[TI * SS];
    __shared__ float red[TI * 16];
    __shared__ float rowmax[TI];
    __shared__ float rowsum[TI];

    const int h    = blockIdx.y;
    const int i0   = blockIdx.x * TI;
    const int t    = threadIdx.x;
    const int lane = t & 31;
    const int wid  = t >> 5;

    const float* Qh  = q   + (size_t)h * LSEQ * DHEAD;
    const float* Kh  = kk_ + (size_t)h * LSEQ * DHEAD;
    const float* Vh  = v   + (size_t)h * LSEQ * DHEAD;
    const float* Rkh = rk  + (size_t)h * LSEQ * LSEQ * DHEAD;
    const float* Rvh = rv  + (size_t)h * LSEQ * LSEQ * DHEAD;

    // ---- stage Q tile (16 x 64) into LDS ----
    {
        const int m  = t >> 4;            // 0..15
        const int kq = (t & 15) * 4;      // 0..60
        const float4 qv = *(const float4*)(Qh + (size_t)(i0 + m) * DHEAD + kq);
        Qs[m * QS + kq + 0] = qv.x;
        Qs[m * QS + kq + 1] = qv.y;
        Qs[m * QS + kq + 2] = qv.z;
        Qs[m * QS + kq + 3] = qv.w;
    }
    __syncthreads();

    // ---- scores: WMMA QK^T + VALU relation_k, scale, mask ----
    const int m16 = lane & 15;
    const int hi  = lane >> 4;

    // A fragment (16x4 fp32 per WMMA step): VGPR0 = K{0,2}, VGPR1 = K{1,3}
    v2f afrag[16];
    #pragma unroll
    for (int kc = 0; kc < 16; ++kc) {
        afrag[kc].x = Qs[m16 * QS + kc * 4 + 2 * hi + 0];
        afrag[kc].y = Qs[m16 * QS + kc * 4 + 2 * hi + 1];
    }

    for (int jt = wid; jt < LSEQ / 16; jt += 8) {   // 4 j-tiles per wave, uniform EXEC
        const int j0 = jt * 16;

        // Stage ALL 16 B-fragments first so the b64 loads issue as a clause
        // (prevents single-buffer load->wait->wmma serialization).
        v2f bfrag[16];
        #pragma unroll
        for (int kc = 0; kc < 16; ++kc) {
            // B[k][n] = K[j0+n][k], n = lane&15, k = 4*kc + 2*hi (+1)
            bfrag[kc] = *(const v2f*)(Kh + (size_t)(j0 + m16) * DHEAD + kc * 4 + 2 * hi);
        }
        v8f c = {};
        #pragma unroll
        for (int kc = 0; kc < 16; ++kc) {
            c = __builtin_amdgcn_wmma_f32_16x16x4_f32(false, afrag[kc], false, bfrag[kc],
                                                      (short)0, c, false, false);
        }
        __builtin_amdgcn_sched_barrier(0);

        // relation_k contribution per C-fragment element, streamed as float4.
        // sched_barrier between r-iterations caps in-flight loads (VGPR pressure).
        #pragma unroll
        for (int r = 0; r < 8; ++r) {
            const int m  = r + 8 * hi;
            const int gi = i0 + m;
            const int gj = j0 + m16;
            const float4* rkp = (const float4*)(Rkh + ((size_t)gi * LSEQ + gj) * DHEAD);
            // prefetch the same (m,n) row of this wave's NEXT j-tile (gj+128)
            if (jt + 8 < LSEQ / 16) {
                __builtin_prefetch(Rkh + ((size_t)gi * LSEQ + gj + 128) * DHEAD, 0, 0);
            }
            float rel0 = 0.f, rel1 = 0.f;
            #pragma unroll
            for (int kq = 0; kq < 16; kq += 2) {
                const float4 qa = *(const float4*)&Qs[m * QS + kq * 4];
                const float4 qb = *(const float4*)&Qs[m * QS + kq * 4 + 4];
                const float4 ra = rkp[kq];
                const float4 rb = rkp[kq + 1];
                rel0 = fmaf(qa.x, ra.x, rel0); rel0 = fmaf(qa.y, ra.y, rel0);
                rel0 = fmaf(qa.z, ra.z, rel0); rel0 = fmaf(qa.w, ra.w, rel0);
                rel1 = fmaf(qb.x, rb.x, rel1); rel1 = fmaf(qb.y, rb.y, rel1);
                rel1 = fmaf(qb.z, rb.z, rel1); rel1 = fmaf(qb.w, rb.w, rel1);
            }
            float s = (c[r] + rel0 + rel1) * 0.125f;     // 1/sqrt(64)
            if (mask[(size_t)gi * LSEQ + gj] == 0) s = -1e9f;
            Sc[m * SS + j0 + m16] = s;
            __builtin_amdgcn_sched_barrier(0);
        }
    }
    __syncthreads();

    // ---- softmax over j (per-row, LDS reductions) ----
    const int row = t >> 4;
    const int sub = t & 15;
    {
        float lmax = -3.0e38f;
        for (int qj = 0; qj < 32; ++qj)
            lmax = fmaxf(lmax, Sc[row * SS + sub * 32 + qj]);
        red[row * 16 + sub] = lmax;
    }
    __syncthreads();
    if (sub == 0) {
        float mx = red[row * 16];
        for (int u = 1; u < 16; ++u) mx = fmaxf(mx, red[row * 16 + u]);
        rowmax[row] = mx;
    }
    __syncthreads();
    {
        const float mx = rowmax[row];
        float lsum = 0.f;
        for (int qj = 0; qj < 32; ++qj) {
            const int j = sub * 32 + qj;
            const float e = __expf(Sc[row * SS + j] - mx);
            Sc[row * SS + j] = e;
            lsum += e;
        }
        red[row * 16 + sub] = lsum;
    }
    __syncthreads();
    if (sub == 0) {
        float sm = 0.f;
        for (int u = 0; u < 16; ++u) sm += red[row * 16 + u];
        rowsum[row] = sm;
    }
    __syncthreads();
    {
        const float inv = 1.0f / rowsum[row];
        float* po = out_p + ((size_t)h * LSEQ + i0 + row) * LSEQ;
        for (int qj = 0; qj < 32; ++qj) {
            const int j = sub * 32 + qj;
            const float p = Sc[row * SS + j] * inv;
            Sc[row * SS + j] = p;
            po[j] = p;
        }
    }
    __syncthreads();

    // ---- output: P@V + sum_j P * relation_v (coalesced float4 streams) ----
    {
        const int m  = t >> 4;
        const int k4 = (t & 15) * 4;
        const int gi = i0 + m;
        float4 acc = {0.f, 0.f, 0.f, 0.f};
        const float* vp  = Vh + k4;
        const float* rvp = Rvh + (size_t)gi * LSEQ * DHEAD + k4;
        #pragma unroll 4
        for (int j = 0; j < LSEQ; ++j) {
            const float  p  = Sc[m * SS + j];
            const float4 vv = *(const float4*)(vp  + (size_t)j * DHEAD);
            const float4 rr = *(const float4*)(rvp + (size_t)j * DHEAD);
            acc.x = fmaf(p, vv.x + rr.x, acc.x);
            acc.y = fmaf(p, vv.y + rr.y, acc.y);
            acc.z = fmaf(p, vv.z + rr.z, acc.z);
            acc.w = fmaf(p, vv.w + rr.w, acc.w);
        }
        *(float4*)(out_attn + ((size_t)h * LSEQ + gi) * DHEAD + k4) = acc;
    }
}

extern "C" void kernel_launch(void* const* d_in, const int* in_sizes, int n_in,
                              void* d_out, int out_size, void* d_ws, size_t ws_size,
                              hipStream_t stream) {
    const float* q    = (const float*)d_in[0];
    const float* k    = (const float*)d_in[1];
    const float* v    = (const float*)d_in[2];
    const float* rk   = (const float*)d_in[3];
    const float* rv   = (const float*)d_in[4];
    const int*   mask = (const int*)d_in[5];

    float* out_attn = (float*)d_out;                              // (1,8,512,64)
    float* out_p    = out_attn + (size_t)NH * LSEQ * DHEAD;       // (1,8,512,512)

    dim3 grid(LSEQ / TI, NH);   // 32 x 8 = 256 workgroups
    ra_attn_kernel<<<grid, 256, 0, stream>>>(q, k, v, rk, rv, mask, out_attn, out_p);
}